// SpectralMoE_51445118272195
// MI455X (gfx1250) — compile-verified
//
#include <hip/hip_runtime.h>
#include <math.h>

typedef __attribute__((ext_vector_type(2))) float v2f;
typedef __attribute__((ext_vector_type(8))) float v8f;

#define CDIM 96
#define HDIM 128
#define WDIM 128
#define BDIM 16
#define EDIM 4
#define TILE 16
#define HSTR 98    // LDS stride (floats) for h[pixel][ci]
#define ASTR 100   // LDS stride (floats) for pwA[co][ci]

// ---------------- Pass 1: global average pool  pooled[b*C+c] ----------------
__global__ __launch_bounds__(256) void pool_kernel(const float* __restrict__ x,
                                                   float* __restrict__ pooled) {
    __shared__ float red[256];
    const int bc = blockIdx.x;                       // b*C + c
    const float* xp = x + (size_t)bc * (HDIM * WDIM);
    float s = 0.f;
    for (int i = threadIdx.x; i < HDIM * WDIM; i += 256) s += xp[i];
    red[threadIdx.x] = s;
    __syncthreads();
    for (int st = 128; st > 0; st >>= 1) {
        if (threadIdx.x < st) red[threadIdx.x] += red[threadIdx.x + st];
        __syncthreads();
    }
    if (threadIdx.x == 0) pooled[bc] = red[0] * (1.0f / (HDIM * WDIM));
}

// ---------------- Pass 2: router (softmax + top-1) --------------------------
__global__ void router_kernel(const float* __restrict__ pooled,
                              const float* __restrict__ rw,
                              const float* __restrict__ rb,
                              float* __restrict__ gate, int* __restrict__ eidx) {
    const int b = threadIdx.x;
    if (b >= BDIM) return;
    float logits[EDIM];
    for (int e = 0; e < EDIM; ++e) {
        float s = rb[e];
        for (int c = 0; c < CDIM; ++c) s = fmaf(pooled[b * CDIM + c], rw[e * CDIM + c], s);
        logits[e] = s;
    }
    float mx = logits[0];
    for (int e = 1; e < EDIM; ++e) mx = fmaxf(mx, logits[e]);
    float w[EDIM], sum = 0.f;
    for (int e = 0; e < EDIM; ++e) { w[e] = expf(logits[e] - mx); sum += w[e]; }
    int best = 0; float bw = w[0];
    for (int e = 1; e < EDIM; ++e) if (w[e] > bw) { bw = w[e]; best = e; }
    const float tw = bw / sum;          // softmax top-1 weight
    gate[b] = tw / (tw + 1e-8f);        // top-k renormalization (k=1)
    eidx[b] = best;
}

// ---------------- Pass 3: fused expert (dw3x3 + GELU + 1x1 via WMMA) --------
__global__ __launch_bounds__(256) void moe_main_kernel(
    const float* __restrict__ x,
    const float* __restrict__ dw_w, const float* __restrict__ dw_b,
    const float* __restrict__ pw_w, const float* __restrict__ pw_b,
    const float* __restrict__ gate, const int* __restrict__ eidx,
    float* __restrict__ out) {
    extern __shared__ float smem[];
    float* hbuf = smem;                        // [256][HSTR]  post-GELU activations
    float* pwA  = hbuf + 256 * HSTR;           // [96][ASTR]   1x1 weights (A matrix)
    float* dww  = pwA + CDIM * ASTR;           // [96*9]
    float* dwb  = dww + CDIM * 9;              // [96]
    float* pwb  = dwb + CDIM;                  // [96]

    const int b   = blockIdx.z;
    const int y0  = blockIdx.y * TILE;
    const int x0  = blockIdx.x * TILE;
    const int tid = threadIdx.x;
    const int e   = eidx[b];
    const float g = gate[b];

    // ---- preload selected expert's weights into LDS ----
    {
        const float* dws = dw_w + (size_t)e * CDIM * 9;
        for (int i = tid; i < CDIM * 9; i += 256) dww[i] = dws[i];
        for (int i = tid; i < CDIM; i += 256) {
            dwb[i] = dw_b[e * CDIM + i];
            pwb[i] = pw_b[e * CDIM + i];
        }
        const float* pws = pw_w + (size_t)e * CDIM * CDIM;
        for (int i = tid; i < CDIM * CDIM; i += 256) {
            const int co = i / CDIM, ci = i - co * CDIM;
            pwA[co * ASTR + ci] = pws[i];
        }
    }
    __syncthreads();

    // ---- Phase A: depthwise 3x3 + bias + exact GELU -> hbuf[p][ci] ----
    {
        const int px = tid & (TILE - 1), py = tid >> 4;
        const int gy = y0 + py, gx = x0 + px;
        const float* xb = x + (size_t)b * CDIM * HDIM * WDIM;
        for (int ci = 0; ci < CDIM; ++ci) {
            const float* xp = xb + (size_t)ci * HDIM * WDIM;
            const float* wk = dww + ci * 9;
            float acc = dwb[ci];
            #pragma unroll
            for (int ky = 0; ky < 3; ++ky) {
                const int yy = gy + ky - 1;
                if (yy < 0 || yy >= HDIM) continue;
                #pragma unroll
                for (int kx = 0; kx < 3; ++kx) {
                    const int xx = gx + kx - 1;
                    if (xx < 0 || xx >= WDIM) continue;
                    acc = fmaf(wk[ky * 3 + kx], xp[yy * WDIM + xx], acc);
                }
            }
            // exact (erf) GELU, matching torch/jax approximate=False
            const float gel = 0.5f * acc * (1.0f + erff(acc * 0.70710678118654752f));
            hbuf[tid * HSTR + ci] = gel;
        }
    }
    __syncthreads();

    // ---- Phase B: out[co,p] = sum_ci pwA[co,ci] * h[ci,p]  (WMMA f32 16x16x4)
    {
        const int wave = tid >> 5;
        const int lane = tid & 31;
        const int n    = lane & 15;   // A: row M index / B: col N index
        const int hi   = lane >> 4;   // selects K pair (0,1) vs (2,3)
        // 6 M-tiles x 16 N-tiles = 96 tiles over 8 waves, 12 consecutive tiles
        // per wave; consecutive tiles share the same M-tile (A operand reuse).
        for (int t = wave * 12; t < wave * 12 + 12; ++t) {
            const int mt  = t >> 4;       // M tile (co)
            const int nt  = t & 15;       // N tile (pixels)
            const int cob = mt * 16;
            const int pb  = nt * 16;
            const float* arow = pwA  + (size_t)(cob + n) * ASTR;  // A[m][k]
            const float* brow = hbuf + (size_t)(pb  + n) * HSTR;  // B[k][n] (stored [n][k])
            v8f acc = {};
            #pragma unroll
            for (int k0 = 0; k0 < CDIM; k0 += 4) {
                const int kk = k0 + 2 * hi;                 // even -> 8B aligned
                const v2f a  = *(const v2f*)(arow + kk);
                const v2f bv = *(const v2f*)(brow + kk);
                acc = __builtin_amdgcn_wmma_f32_16x16x4_f32(
                    false, a, false, bv, (short)0, acc, false, false);
            }
            // C/D layout: VGPR r -> M = r + 8*hi, N = lane&15
            const int p  = pb + n;
            const int py = p >> 4, px = p & 15;
            float* op = out + (((size_t)b * CDIM) * HDIM + (y0 + py)) * WDIM + (x0 + px);
            #pragma unroll
            for (int r = 0; r < 8; ++r) {
                const int co = cob + 8 * hi + r;
                op[(size_t)co * HDIM * WDIM] = g * (acc[r] + pwb[co]);
            }
        }
    }
}

extern "C" void kernel_launch(void* const* d_in, const int* in_sizes, int n_in,
                              void* d_out, int out_size, void* d_ws, size_t ws_size,
                              hipStream_t stream) {
    const float* x    = (const float*)d_in[0];
    const float* dw_w = (const float*)d_in[1];
    const float* dw_b = (const float*)d_in[2];
    const float* pw_w = (const float*)d_in[3];
    const float* pw_b = (const float*)d_in[4];
    const float* rw   = (const float*)d_in[5];
    const float* rb   = (const float*)d_in[6];
    float* out = (float*)d_out;

    float* pooled = (float*)d_ws;            // [B*C]
    float* gate   = pooled + BDIM * CDIM;    // [B]
    int*   eidx   = (int*)(gate + BDIM);     // [B]

    pool_kernel<<<BDIM * CDIM, 256, 0, stream>>>(x, pooled);
    router_kernel<<<1, 32, 0, stream>>>(pooled, rw, rb, gate, eidx);

    const size_t shmem =
        (size_t)(256 * HSTR + CDIM * ASTR + CDIM * 9 + CDIM + CDIM) * sizeof(float);
    // Unconditional + idempotent (no static guards; graph-capture safe).
    hipFuncSetAttribute((const void*)moe_main_kernel,
                        hipFuncAttributeMaxDynamicSharedMemorySize, (int)shmem);
    dim3 grid(WDIM / TILE, HDIM / TILE, BDIM);
    moe_main_kernel<<<grid, 256, shmem, stream>>>(x, dw_w, dw_b, pw_w, pw_b,
                                                  gate, eidx, out);
}